// STKG_PLM_76501957476484
// MI455X (gfx1250) — compile-verified
//
#include <hip/hip_runtime.h>
#include <math.h>

typedef float v2f __attribute__((ext_vector_type(2)));
typedef float v8f __attribute__((ext_vector_type(8)));

#define NK 32     // neighbors per node
#define DIM 128   // embedding dim

// One workgroup (256 threads = 8 wave32) per node. Fuses both GAT layers.
__launch_bounds__(256)
__global__ void kgat_fused_kernel(const float* __restrict__ src,
                                  const float* __restrict__ dst,
                                  const float* __restrict__ rel,
                                  const float* __restrict__ W,   // [256,128] row-major
                                  const float* __restrict__ b,   // [128]
                                  const int*   __restrict__ adj, // [N,32]
                                  float* __restrict__ out) {
  __shared__ float sDst[NK][DIM];   // 16 KB
  __shared__ float sRel[NK][DIM];   // 16 KB
  __shared__ float sH[NK][DIM];     // 16 KB : dst @ W_bot + b (layer-invariant)
  __shared__ float sSrc[DIM];
  __shared__ float sSrcW[DIM];
  __shared__ float sDotHR[NK];
  __shared__ float sHn[NK];
  __shared__ float sRn[NK];
  __shared__ float sE[NK];
  __shared__ float sAtt[NK];
  __shared__ int   sAdj[NK];

  const int n    = blockIdx.x;
  const int t    = threadIdx.x;
  const int lane = t & 31;
  const int w    = t >> 5;

  // ---------- Phase 0: stage node data into LDS (coalesced float4) ----------
  {
    const float4* d4 = (const float4*)(dst + (size_t)n * NK * DIM);
    const float4* r4 = (const float4*)(rel + (size_t)n * NK * DIM);
    float4* sD4 = (float4*)&sDst[0][0];
    float4* sR4 = (float4*)&sRel[0][0];
    for (int i = t; i < NK * DIM / 4; i += 256) { sD4[i] = d4[i]; sR4[i] = r4[i]; }
    if (t < DIM) sSrc[t] = src[(size_t)n * DIM + t];
    if (t < NK)  sAdj[t] = adj[(size_t)n * NK + t];
  }
  __syncthreads();

  // ---------- Phase 1: H = dst @ W_bot + b  (f32 WMMA 16x16x4) ----------
  // Wave w computes H rows [ (w/4)*16 .. +16 ), cols [ (w%4)*32 .. +32 ) as two 16x16 tiles.
  {
    const float* Wb   = W + 128 * DIM;          // W_bot = W[128:256, :]
    const int mrow0   = (w >> 2) * 16;
    const int nb      = (w & 3) * 32;
    const int l15     = lane & 15;
    const int khalf   = (lane >> 4) * 2;        // lanes 0-15 -> K0/K1, lanes 16-31 -> K2/K3
    v8f acc0 = {};
    v8f acc1 = {};
    for (int kt = 0; kt < 32; ++kt) {
      const int k0 = kt * 4;
      // A operand: 16x4 f32 tile of dst (ISA A layout)
      v2f a;
      a.x = sDst[mrow0 + l15][k0 + khalf];
      a.y = sDst[mrow0 + l15][k0 + khalf + 1];
      // B operands: 4x16 f32 tiles of W_bot (row-striped like C/D: V0 = K rows 0/2, V1 = rows 1/3)
      v2f b0, b1;
      b0.x = Wb[(k0 + khalf)     * DIM + nb + l15];
      b0.y = Wb[(k0 + khalf + 1) * DIM + nb + l15];
      b1.x = Wb[(k0 + khalf)     * DIM + nb + 16 + l15];
      b1.y = Wb[(k0 + khalf + 1) * DIM + nb + 16 + l15];
      acc0 = __builtin_amdgcn_wmma_f32_16x16x4_f32(false, a, false, b0, (short)0, acc0, false, false);
      acc1 = __builtin_amdgcn_wmma_f32_16x16x4_f32(false, a, false, b1, (short)0, acc1, false, false);
    }
    // C/D layout: VGPR j -> row (j + 8*(lane>=16)), col = lane&15 within tile. Add bias here.
    const int rbase = mrow0 + ((lane >> 4) ? 8 : 0);
    for (int j = 0; j < 8; ++j) {
      const int r  = rbase + j;
      const int c0 = nb + l15;
      const int c1 = nb + 16 + l15;
      sH[r][c0] = acc0[j] + b[c0];
      sH[r][c1] = acc1[j] + b[c1];
    }
  }
  __syncthreads();

  // ---------- Phase 2: layer-invariant reductions: H.rel, |H|^2, |rel|^2 ----------
  for (int kk = 0; kk < 4; ++kk) {
    const int k = w * 4 + kk;
    float phr = 0.f, phh = 0.f, prr = 0.f;
    for (int dd = 0; dd < 4; ++dd) {
      const int d = lane + dd * 32;
      const float hv = sH[k][d], rv = sRel[k][d];
      phr = fmaf(hv, rv, phr);
      phh = fmaf(hv, hv, phh);
      prr = fmaf(rv, rv, prr);
    }
    for (int off = 16; off; off >>= 1) {
      phr += __shfl_xor(phr, off, 32);
      phh += __shfl_xor(phh, off, 32);
      prr += __shfl_xor(prr, off, 32);
    }
    if (lane == 0) { sDotHR[k] = phr; sHn[k] = phh; sRn[k] = prr; }
  }
  __syncthreads();

  // ---------- Phase 3: two GAT layers (only src changes between them) ----------
  for (int layer = 0; layer < 2; ++layer) {
    // srcW = cur_src @ W_top  (coalesced reads of W rows, L2-resident)
    if (t < DIM) {
      float a = 0.f;
      for (int d = 0; d < DIM; ++d) a = fmaf(sSrc[d], W[d * DIM + t], a);
      sSrcW[t] = a;
    }
    __syncthreads();

    // attention logits: e_k = cos(srcW + H_k, rel_k) -> LeakyReLU -> mask
    {
      float s2p = 0.f;
      for (int dd = 0; dd < 4; ++dd) {
        const float v = sSrcW[lane + dd * 32];
        s2p = fmaf(v, v, s2p);
      }
      for (int off = 16; off; off >>= 1) s2p += __shfl_xor(s2p, off, 32);

      for (int kk = 0; kk < 4; ++kk) {
        const int k = w * 4 + kk;
        float psr = 0.f, psh = 0.f;
        for (int dd = 0; dd < 4; ++dd) {
          const int d = lane + dd * 32;
          const float sw = sSrcW[d];
          psr = fmaf(sw, sRel[k][d], psr);
          psh = fmaf(sw, sH[k][d],  psh);
        }
        for (int off = 16; off; off >>= 1) {
          psr += __shfl_xor(psr, off, 32);
          psh += __shfl_xor(psh, off, 32);
        }
        if (lane == 0) {
          const float num = psr + sDotHR[k];
          const float hh  = s2p + 2.f * psh + sHn[k];
          const float den = fmaxf(sqrtf(hh), 1e-8f) * fmaxf(sqrtf(sRn[k]), 1e-8f);
          float e = num / den;
          e = (e > 0.f) ? e : 0.2f * e;          // LeakyReLU(0.2)
          e = (sAdj[k] > 0) ? e : -9e15f;        // mask
          sE[k] = e;
        }
      }
    }
    __syncthreads();

    // softmax over K=32 (single wave)
    if (w == 0) {
      const float e = sE[lane];
      float m = e;
      for (int off = 16; off; off >>= 1) m = fmaxf(m, __shfl_xor(m, off, 32));
      const float p = __expf(e - m);
      float s = p;
      for (int off = 16; off; off >>= 1) s += __shfl_xor(s, off, 32);
      sAtt[lane] = p / s;
    }
    __syncthreads();

    // aggregate + residual: src <- att @ dst + src
    if (t < DIM) {
      float a = 0.f;
      for (int k = 0; k < NK; ++k) a = fmaf(sAtt[k], sDst[k][t], a);
      sSrc[t] = a + sSrc[t];
    }
    __syncthreads();
  }

  if (t < DIM) out[(size_t)n * DIM + t] = sSrc[t];
}

extern "C" void kernel_launch(void* const* d_in, const int* in_sizes, int n_in,
                              void* d_out, int out_size, void* d_ws, size_t ws_size,
                              hipStream_t stream) {
  const float* src = (const float*)d_in[0];   // [8192,128]
  const float* dst = (const float*)d_in[1];   // [8192,32,128]
  const float* rel = (const float*)d_in[2];   // [8192,32,128]
  const float* W   = (const float*)d_in[3];   // [256,128]
  const float* b   = (const float*)d_in[4];   // [128]
  const int*   adj = (const int*)d_in[5];     // [8192,32]
  float* out = (float*)d_out;                 // [8192,128]

  const int n_nodes = in_sizes[0] / DIM;      // 8192
  kgat_fused_kernel<<<n_nodes, 256, 0, stream>>>(src, dst, rel, W, b, adj, out);
}